// MaxPoolingAggregator_77730318123217
// MI455X (gfx1250) — compile-verified
//
#include <hip/hip_runtime.h>
#include <hip/hip_bf16.h>
#include <math.h>

// Problem constants (from reference): B=2, N=2048, D=128, H=32, N_UNKNOWN=256
#define NN      2048
#define DD      128
#define HH      32
#define BB      2
#define NUNK    256
#define KSPLIT  16            // split the k (neighbor) reduction 16 ways -> 1024 pooling waves

typedef float v2f __attribute__((ext_vector_type(2)));
typedef float v8f __attribute__((ext_vector_type(8)));

// ---------------------------------------------------------------------------
// Mask build: unknown_mask[l] = 1 if l appears in batch_unknown_nodes
// ---------------------------------------------------------------------------
__global__ void zero_mask_kernel(int* __restrict__ mask) {
    int i = blockIdx.x * blockDim.x + threadIdx.x;
    if (i < NN) mask[i] = 0;
}

__global__ void scatter_mask_kernel(const int* __restrict__ unk, int* __restrict__ mask) {
    int i = blockIdx.x * blockDim.x + threadIdx.x;
    if (i < NUNK) mask[unk[i]] = 1;   // all racers write 1 -> deterministic
}

// ---------------------------------------------------------------------------
// f32 WMMA GEMM: out[M x 32] = A[M x K] * Bm[K x 32] (+bias, ReLU if BIAS_RELU)
// One full wave32 per 16x16 output tile (EXEC all ones, as WMMA requires).
// Uses V_WMMA_F32_16X16X4_F32 so the math stays exact fp32 like the reference.
//
// Fragment layouts (ISA 7.12.2):
//   A 16x4 : a.x = A[m=lane&15][k + 2*(lane>>4)],  a.y = A[m][k + 2*(lane>>4) + 1]
//   B 4x16 : b.x = Bm[k + 2*(lane>>4)][n0 + (lane&15)], b.y = next K row
//   C 16x16: c[r] = C[m = r + 8*(lane>>4)][n0 + (lane&15)]
// ---------------------------------------------------------------------------
template <int K, bool BIAS_RELU>
__global__ void gemm_wmma_f32_kernel(const float* __restrict__ A,
                                     const float* __restrict__ Bm,
                                     const float* __restrict__ bias,
                                     float* __restrict__ out) {
    const int lane  = threadIdx.x;           // 0..31
    const int ncol  = lane & 15;             // column within 16-wide N tile / row M within A
    const int khalf = lane >> 4;             // 0 or 1: selects K sub-pair
    const int tile  = blockIdx.x;            // (M/16)*2 tiles total (HH==32 -> 2 N tiles)
    const int m0    = (tile >> 1) * 16;
    const int n0    = (tile & 1) * 16;

    v8f c = {};
    const float* arow = A + (size_t)(m0 + ncol) * K + 2 * khalf;
#pragma unroll
    for (int k = 0; k < K; k += 4) {
        v2f a, b;
        a.x = arow[k];                       // A[m][k + 2*khalf]
        a.y = arow[k + 1];                   // A[m][k + 2*khalf + 1]
        const float* bp = Bm + (size_t)(k + 2 * khalf) * HH + n0 + ncol;
        b.x = bp[0];                         // Bm[k + 2*khalf    ][n]
        b.y = bp[HH];                        // Bm[k + 2*khalf + 1][n]
        // (neg_a, A, neg_b, B, c_mod, C, reuse_a, reuse_b)
        c = __builtin_amdgcn_wmma_f32_16x16x4_f32(false, a, false, b,
                                                  (short)0, c, false, false);
    }

    float bv = 0.0f;
    if (BIAS_RELU) bv = bias[n0 + ncol];
#pragma unroll
    for (int r = 0; r < 8; ++r) {
        float v = c[r];
        if (BIAS_RELU) v = fmaxf(v + bv, 0.0f);
        out[(size_t)(m0 + r + 8 * khalf) * HH + n0 + ncol] = v;
    }
}

// ---------------------------------------------------------------------------
// Masked max-pool, k-split partials.
// Wave (32 lanes, lane == h) handles one 32-wide l-tile and one k-chunk of
// NN/KSPLIT neighbors. adj row chunks are coalesced 128B loads; the per-l
// adjacency bit is wave-uniform inside the unrolled l loop, so __shfl lowers
// to v_readlane. contribution = xw * a  (a in {0,1}) exactly as the reference.
// part[s][b][l][h]
// ---------------------------------------------------------------------------
__global__ void pool_partial_kernel(const float* __restrict__ adj,
                                    const int*   __restrict__ unk_mask,
                                    const float* __restrict__ xw,
                                    float*       __restrict__ part) {
    const int lane = threadIdx.x;            // 0..31  (h for xw, l-offset for adj)
    const int l0   = blockIdx.x * 32;        // 64 l-tiles
    const int s    = blockIdx.y;             // 0..KSPLIT-1
    const int kcnt = NN / KSPLIT;
    const int k0   = s * kcnt;

    const float colOK = unk_mask[l0 + lane] ? 0.0f : 1.0f;

    float acc0[32], acc1[32];
#pragma unroll
    for (int l = 0; l < 32; ++l) { acc0[l] = -INFINITY; acc1[l] = -INFINITY; }

    for (int k = k0; k < k0 + kcnt; ++k) {
        // prefetch next adj row chunk (global_prefetch_b8), clamped in-buffer
        int kpf = (k + 8 < NN) ? (k + 8) : k;
        __builtin_prefetch(&adj[(size_t)kpf * NN + l0 + lane], 0, 3);

        float av = adj[(size_t)k * NN + l0 + lane];
        av = (av > 0.0f) ? colOK : 0.0f;               // binarize + column mask
        const float x0 = xw[(size_t)k * HH + lane];            // b = 0
        const float x1 = xw[(size_t)(NN + k) * HH + lane];     // b = 1
#pragma unroll
        for (int l = 0; l < 32; ++l) {
            float al = __shfl(av, l, 32);              // broadcast a[k, l0+l]
            acc0[l] = fmaxf(acc0[l], x0 * al);
            acc1[l] = fmaxf(acc1[l], x1 * al);
        }
    }

#pragma unroll
    for (int l = 0; l < 32; ++l) {
        part[(((size_t)s * BB + 0) * NN + l0 + l) * HH + lane] = acc0[l];
        part[(((size_t)s * BB + 1) * NN + l0 + l) * HH + lane] = acc1[l];
    }
}

// ---------------------------------------------------------------------------
// Reduce the KSPLIT partial maxima -> pooled[b][l][h]
// ---------------------------------------------------------------------------
__global__ void pool_reduce_kernel(const float* __restrict__ part,
                                   float* __restrict__ pooled) {
    int i = blockIdx.x * blockDim.x + threadIdx.x;     // over BB*NN*HH = 131072
    if (i >= BB * NN * HH) return;
    const int h = i & (HH - 1);
    const int l = (i >> 5) & (NN - 1);
    const int b = i >> 16;                             // NN*HH == 65536
    float m = -INFINITY;
#pragma unroll
    for (int s = 0; s < KSPLIT; ++s)
        m = fmaxf(m, part[(((size_t)s * BB + b) * NN + l) * HH + h]);
    pooled[i] = m;
}

// ---------------------------------------------------------------------------
// Launch
// inputs: 0=adj [N*N] f32, 1=x [B*N*D] f32, 2=batch_unknown_nodes [256] i32,
//         3=W1 [D*H] f32, 4=W2 [H*H] f32, 5=b2 [H] f32
// out: [B*N*H] f32
// ---------------------------------------------------------------------------
extern "C" void kernel_launch(void* const* d_in, const int* in_sizes, int n_in,
                              void* d_out, int out_size, void* d_ws, size_t ws_size,
                              hipStream_t stream) {
    const float* adj = (const float*)d_in[0];
    const float* x   = (const float*)d_in[1];
    const int*   unk = (const int*)  d_in[2];
    const float* W1  = (const float*)d_in[3];
    const float* W2  = (const float*)d_in[4];
    const float* b2  = (const float*)d_in[5];
    float* out = (float*)d_out;

    // workspace layout (bytes): xw 512KB | pooled 512KB | mask 8KB(+pad) | partials 8MB
    char* ws = (char*)d_ws;
    float* xw     = (float*)(ws);                                  // B*N*H
    float* pooled = (float*)(ws + (size_t)BB * NN * HH * 4);       // B*N*H
    int*   mask   = (int*)  (ws + (size_t)2 * BB * NN * HH * 4);   // N
    float* part   = (float*)(ws + (size_t)2 * BB * NN * HH * 4 + 64 * 1024); // KSPLIT*B*N*H

    // 1) build unknown-column mask
    zero_mask_kernel<<<(NN + 255) / 256, 256, 0, stream>>>(mask);
    scatter_mask_kernel<<<(NUNK + 255) / 256, 256, 0, stream>>>(unk, mask);

    // 2) fc1: xw = x @ W1   ([B*N, D] @ [D, H]); 256 M-tiles * 2 N-tiles waves
    gemm_wmma_f32_kernel<DD, false><<<(BB * NN / 16) * 2, 32, 0, stream>>>(x, W1, nullptr, xw);

    // 3) masked max-pool partials: 64 l-tiles x KSPLIT k-chunks, 1 wave each
    pool_partial_kernel<<<dim3(NN / 32, KSPLIT), 32, 0, stream>>>(adj, mask, xw, part);

    // 4) reduce partials
    pool_reduce_kernel<<<(BB * NN * HH + 255) / 256, 256, 0, stream>>>(part, pooled);

    // 5) fc2 + bias + ReLU: out = relu(pooled @ W2 + b2)
    gemm_wmma_f32_kernel<HH, true><<<(BB * NN / 16) * 2, 32, 0, stream>>>(pooled, W2, b2, out);
}